// TransformerConv_53644141527046
// MI455X (gfx1250) — compile-verified
//
#include <hip/hip_runtime.h>
#include <hip/hip_bf16.h>
#include <math.h>

#define N_NODES 50000
#define E_EDGES 800000
#define NIN 128
#define HC 128
#define HEADS 4
#define ED 44
#define ELDS_STRIDE 129   // 16x128 tile padded to kill LDS bank conflicts

typedef float v2f __attribute__((ext_vector_type(2)));
typedef float v8f __attribute__((ext_vector_type(8)));

__device__ __forceinline__ v8f wmma_f32x4(v2f a, v2f b, v8f c) {
  // D = A(16x4) * B(4x16) + C(16x16), fp32, wave32
  return __builtin_amdgcn_wmma_f32_16x16x4_f32(
      /*neg_a=*/false, a, /*neg_b=*/false, b,
      /*c_mod=*/(short)0, c, /*reuse_a=*/false, /*reuse_b=*/false);
}

// ---------------------------------------------------------------- init
__global__ void tc_init_kernel(int* __restrict__ m_enc, float* __restrict__ denom, int count) {
  int i = blockIdx.x * blockDim.x + threadIdx.x;
  if (i < count) {
    m_enc[i] = (int)0x80000000;  // below any ordered-encoded float
    denom[i] = 0.0f;
  }
}

// ---------------------------------------------------------------- node GEMMs
// One wave = one 16x16 output tile; A fragment shared across Wq/Wk/Wv/Wskip.
__global__ __launch_bounds__(256)
void tc_node_gemm_kernel(const float* __restrict__ x,
                         const float* __restrict__ Wq, const float* __restrict__ bq,
                         const float* __restrict__ Wk, const float* __restrict__ bk,
                         const float* __restrict__ Wv, const float* __restrict__ bv,
                         const float* __restrict__ Ws, const float* __restrict__ bs,
                         float* __restrict__ Q, float* __restrict__ K,
                         float* __restrict__ V, float* __restrict__ OutSkip) {
  const int wave  = threadIdx.x >> 5;          // 0..7 -> column tile
  const int lane  = threadIdx.x & 31;
  const int mloc  = lane & 15;
  const int khalf = lane >> 4;                 // 0: K=0,1  1: K=2,3
  const int row0  = blockIdx.x * 16;
  const int n     = wave * 16 + (lane & 15);

  const float* xrow = x + (size_t)(row0 + mloc) * NIN + 2 * khalf;

  v8f cq = {}; v8f ck = {}; v8f cv = {}; v8f cs = {};
  const float bqv = bq[n], bkv = bk[n], bvv = bv[n], bsv = bs[n];

  #pragma unroll 4
  for (int kk = 0; kk < NIN; kk += 4) {
    v2f a = *(const v2f*)(xrow + kk);
    const int r0 = (kk + 2 * khalf) * HC + n;
    v2f wq2 = { Wq[r0], Wq[r0 + HC] };
    v2f wk2 = { Wk[r0], Wk[r0 + HC] };
    v2f wv2 = { Wv[r0], Wv[r0 + HC] };
    v2f ws2 = { Ws[r0], Ws[r0 + HC] };
    cq = wmma_f32x4(a, wq2, cq);
    ck = wmma_f32x4(a, wk2, ck);
    cv = wmma_f32x4(a, wv2, cv);
    cs = wmma_f32x4(a, ws2, cs);
  }

  const float qscale = 0.17677669529663688f;   // 1/sqrt(C=32), folded into Q
  #pragma unroll
  for (int r = 0; r < 8; ++r) {
    size_t o = (size_t)(row0 + r + 8 * khalf) * HC + n;
    Q[o]       = (cq[r] + bqv) * qscale;
    K[o]       = ck[r] + bkv;
    V[o]       = cv[r] + bvv;
    OutSkip[o] = cs[r] + bsv;                  // output pre-loaded with root skip
  }
}

// ---------------------------------------------------------------- e-tile GEMM
// 16 edges x 128 cols = edge_attr[16x44] @ We[44x128], result into LDS.
__device__ __forceinline__ void tc_compute_e_tile(const float* __restrict__ edge_attr,
                                                  const float* __restrict__ We,
                                                  int e0, int lane,
                                                  float* __restrict__ elds) {
  const int mloc  = lane & 15;
  const int khalf = lane >> 4;
  const int nloc  = lane & 15;
  const float* arow = edge_attr + (size_t)(e0 + mloc) * ED + 2 * khalf;

  v2f afrag[11];
  #pragma unroll
  for (int s = 0; s < 11; ++s) afrag[s] = *(const v2f*)(arow + 4 * s);

  #pragma unroll
  for (int j = 0; j < 8; ++j) {
    v8f acc = {};
    #pragma unroll
    for (int s = 0; s < 11; ++s) {
      const int krow = 4 * s + 2 * khalf;
      v2f b = { We[krow * HC + j * 16 + nloc], We[(krow + 1) * HC + j * 16 + nloc] };
      acc = wmma_f32x4(afrag[s], b, acc);
    }
    #pragma unroll
    for (int r = 0; r < 8; ++r)
      elds[(r + 8 * khalf) * ELDS_STRIDE + j * 16 + nloc] = acc[r];
  }
}

// ---------------------------------------------------------------- pass 1: alpha
__global__ __launch_bounds__(128)
void tc_edge_alpha_kernel(const float* __restrict__ edge_attr, const float* __restrict__ We,
                          const long long* __restrict__ ei,
                          const float* __restrict__ Q, const float* __restrict__ K,
                          float* __restrict__ alpha, int* __restrict__ m_enc) {
  __shared__ float elds[4][16 * ELDS_STRIDE];
  const int wave = threadIdx.x >> 5;
  const int lane = threadIdx.x & 31;
  const int e0   = (blockIdx.x * 4 + wave) * 16;
  float* my = elds[wave];

  tc_compute_e_tile(edge_attr, We, e0, lane, my);

  const int eloc = lane & 15;
  const int half = lane >> 4;                   // each lane covers 2 heads
  const int eidx = e0 + eloc;
  const long long s = ei[eidx];
  const long long d = ei[E_EDGES + eidx];

  const float* qrow = Q + (size_t)d * HC + half * 64;   // Q pre-scaled by 1/sqrt(C)
  const float* krow = K + (size_t)s * HC + half * 64;
  const float* erow = my + eloc * ELDS_STRIDE + half * 64;

  float a0 = 0.0f, a1 = 0.0f;
  #pragma unroll 8
  for (int t = 0; t < 32; ++t)  a0 += qrow[t] * (krow[t] + erow[t]);
  #pragma unroll 8
  for (int t = 32; t < 64; ++t) a1 += qrow[t] * (krow[t] + erow[t]);

  const int h0 = half * 2;
  alpha[(size_t)eidx * HEADS + h0]     = a0;
  alpha[(size_t)eidx * HEADS + h0 + 1] = a1;

  int ia0 = __float_as_int(a0); ia0 = (ia0 >= 0) ? ia0 : (ia0 ^ 0x7fffffff);
  int ia1 = __float_as_int(a1); ia1 = (ia1 >= 0) ? ia1 : (ia1 ^ 0x7fffffff);
  atomicMax(&m_enc[(int)d * HEADS + h0],     ia0);
  atomicMax(&m_enc[(int)d * HEADS + h0 + 1], ia1);
}

// ---------------------------------------------------------------- softmax denom
__global__ void tc_denom_kernel(const long long* __restrict__ ei,
                                const int* __restrict__ m_enc,
                                float* __restrict__ alpha, float* __restrict__ denom) {
  long long tid = (long long)blockIdx.x * blockDim.x + threadIdx.x;
  if (tid >= (long long)E_EDGES * HEADS) return;
  const int e = (int)(tid >> 2);
  const int h = (int)(tid & 3);
  const long long d = ei[E_EDGES + e];
  int me = m_enc[(int)d * HEADS + h];
  const float m = __int_as_float((me >= 0) ? me : (me ^ 0x7fffffff));
  const float a = __expf(alpha[tid] - m);
  alpha[tid] = a;                               // keep unnormalized weight
  atomicAdd(&denom[(int)d * HEADS + h], a);
}

// ---------------------------------------------------------------- pass 2: aggregate
__global__ __launch_bounds__(128)
void tc_aggregate_kernel(const float* __restrict__ edge_attr, const float* __restrict__ We,
                         const long long* __restrict__ ei,
                         const float* __restrict__ V, const float* __restrict__ alpha,
                         const float* __restrict__ denom, float* __restrict__ out) {
  __shared__ float elds[4][16 * ELDS_STRIDE];
  const int wave = threadIdx.x >> 5;
  const int lane = threadIdx.x & 31;
  const int e0   = (blockIdx.x * 4 + wave) * 16;
  float* my = elds[wave];

  tc_compute_e_tile(edge_attr, We, e0, lane, my);   // recompute e (cheap vs 410 MB spill)

  const int eloc = lane & 15;
  const int half = lane >> 4;
  const int eidx = e0 + eloc;
  const long long s = ei[eidx];
  const long long d = ei[E_EDGES + eidx];
  const int h0 = half * 2;

  const float w0 = alpha[(size_t)eidx * HEADS + h0]
                   / (denom[(int)d * HEADS + h0] + 1e-16f);
  const float w1 = alpha[(size_t)eidx * HEADS + h0 + 1]
                   / (denom[(int)d * HEADS + h0 + 1] + 1e-16f);

  const float* vrow = V + (size_t)s * HC + half * 64;
  const float* erow = my + eloc * ELDS_STRIDE + half * 64;
  float* orow = out + (size_t)d * HC + half * 64;

  #pragma unroll 8
  for (int t = 0; t < 32; ++t)  atomicAdd(&orow[t], w0 * (vrow[t] + erow[t]));
  #pragma unroll 8
  for (int t = 32; t < 64; ++t) atomicAdd(&orow[t], w1 * (vrow[t] + erow[t]));
}

// ---------------------------------------------------------------- launch
extern "C" void kernel_launch(void* const* d_in, const int* in_sizes, int n_in,
                              void* d_out, int out_size, void* d_ws, size_t ws_size,
                              hipStream_t stream) {
  const float*     x  = (const float*)d_in[0];
  const long long* ei = (const long long*)d_in[1];   // int64 [2, E]
  const float*     ea = (const float*)d_in[2];
  const float*     Wq = (const float*)d_in[3];
  const float*     bq = (const float*)d_in[4];
  const float*     Wk = (const float*)d_in[5];
  const float*     bk = (const float*)d_in[6];
  const float*     Wv = (const float*)d_in[7];
  const float*     bv = (const float*)d_in[8];
  const float*     We = (const float*)d_in[9];
  const float*     Ws = (const float*)d_in[10];
  const float*     bs = (const float*)d_in[11];
  float* out = (float*)d_out;

  // Workspace layout (~91 MB): Q | K | V | alpha | m_enc | denom
  const size_t NHC = (size_t)N_NODES * HC;
  float* Q     = (float*)d_ws;
  float* K     = Q + NHC;
  float* V     = K + NHC;
  float* alpha = V + NHC;
  int*   m_enc = (int*)(alpha + (size_t)E_EDGES * HEADS);
  float* denom = (float*)(m_enc + (size_t)N_NODES * HEADS);

  const int nh = N_NODES * HEADS;
  tc_init_kernel<<<(nh + 255) / 256, 256, 0, stream>>>(m_enc, denom, nh);

  tc_node_gemm_kernel<<<N_NODES / 16, 256, 0, stream>>>(
      x, Wq, bq, Wk, bk, Wv, bv, Ws, bs, Q, K, V, out);

  tc_edge_alpha_kernel<<<E_EDGES / 64, 128, 0, stream>>>(
      ea, We, ei, Q, K, alpha, m_enc);

  const long long ehn = (long long)E_EDGES * HEADS;
  tc_denom_kernel<<<(int)((ehn + 255) / 256), 256, 0, stream>>>(ei, m_enc, alpha, denom);

  tc_aggregate_kernel<<<E_EDGES / 64, 128, 0, stream>>>(
      ea, We, ei, V, alpha, denom, out);
}